// SpikingConv2DLayer_custom_fast_11381663334465
// MI455X (gfx1250) — compile-verified
//
#include <hip/hip_runtime.h>
#include <hip/hip_bf16.h>

// ---------------------------------------------------------------------------
// SpikingConv2DLayer forward on gfx1250 (MI455X).
//
// Math collapse: mem_t = (t+1)*conv, spikes are one-hot at the first t where
// mem_t*inv_norm[c] - bias[c] > 0.  Output = spk_rec (16,32,32,64,64) f32
// followed by loss and spk_spread_loss scalars.
//
// Perf model: 256 MB of one-shot stores dominate (~11 us @ 23.3 TB/s HBM);
// conv is 113 MFLOP (noise).  x (768 KB) is L2-resident; weights staged in LDS.
// Conv done as implicit GEMM with V_WMMA_F32_16X16X4_F32: M=16 channels,
// N=16 pixels, K=27 padded to 28 (7 WMMA steps of K=4) per wave.
// spk_rec stores use the NT temporal hint: the 256 MB stream exceeds the
// 192 MB L2 and is never re-read, so it must not evict the x working set.
// ---------------------------------------------------------------------------

typedef __attribute__((ext_vector_type(2))) float v2f;
typedef __attribute__((ext_vector_type(8))) float v8f;

#define B_IMG     16
#define C_IN      3
#define C_OUT     32
#define IMG_H     64
#define IMG_W     64
#define HW        (IMG_H * IMG_W)              // 4096
#define T_STEPS   32
#define KSIZE     27                           // 3*3*3
#define KPAD      28                           // pad to multiple of 4
#define SPK_ELEMS ((size_t)B_IMG * C_OUT * T_STEPS * HW)   // 67108864
#define PIX_DENOM ((float)(B_IMG * C_OUT * HW))            // 2097152

__global__ __launch_bounds__(256)
void spiking_conv_wmma_kernel(const float* __restrict__ x,
                              const float* __restrict__ w,
                              const float* __restrict__ bias,
                              float* __restrict__ out,
                              unsigned int* __restrict__ gcnt)
{
    // LDS: padded weights [32][28], per-channel 1/(||w||^2+eps), bias, counters
    __shared__ float        wpad[C_OUT][KPAD];
    __shared__ float        invn[C_OUT];
    __shared__ float        bsh[C_OUT];
    __shared__ unsigned int cnt[T_STEPS];

    const int tid = threadIdx.x;
    if (tid < C_OUT) {
        float nrm = 0.f;
        #pragma unroll
        for (int k = 0; k < KSIZE; ++k) {
            float ww = w[tid * KSIZE + k];
            wpad[tid][k] = ww;
            nrm += ww * ww;
        }
        wpad[tid][KSIZE] = 0.f;                  // K=27 pad lane
        invn[tid] = 1.0f / (nrm + 1e-8f);
        bsh[tid]  = bias[tid];
        cnt[tid]  = 0u;
    }
    __syncthreads();

    const int lane    = tid & 31;
    const int wave    = tid >> 5;
    const int gtile   = blockIdx.x * 8 + wave;   // 0..8191 wave-tiles
    const int chBase  = (gtile & 1) * 16;        // channel tile: 0 or 16
    const int pixTile = gtile >> 1;              // 0..4095
    const int n       = lane & 15;               // N index (pixel in tile)
    const int hiHalf  = lane >> 4;               // lane half selects K 0/1 vs 2/3

    // 16 consecutive pixels: 4096%16==0 and 64%16==0, never crosses b or h.
    const int p    = pixTile * 16 + n;
    const int bImg = p >> 12;
    const int h    = (p >> 6) & 63;
    const int wc   = p & 63;

    // ---- implicit GEMM via V_WMMA_F32_16X16X4_F32, 7 K-steps -------------
    v8f acc = {};
    #pragma unroll
    for (int k0 = 0; k0 < KPAD; k0 += 4) {
        const int ka = k0 + hiHalf * 2;          // per-half K offset (ISA A/B split)

        // A (16x4, M=channel): lane's row M = n, K = ka, ka+1 (from LDS, padded)
        v2f A;
        A.x = wpad[chBase + n][ka];
        A.y = wpad[chBase + n][ka + 1];

        // B (4x16, N=pixel): gather x with zero halo padding; K>=27 -> 0
        v2f Bv;
        #pragma unroll
        for (int j = 0; j < 2; ++j) {
            const int k = ka + j;
            float val = 0.f;
            if (k < KSIZE) {
                const int ci = k / 9;
                const int r9 = k - ci * 9;
                const int kh = r9 / 3;
                const int kw = r9 - kh * 3;
                const int hy = h + kh - 1;
                const int wx = wc + kw - 1;
                if (hy >= 0 && hy < IMG_H && wx >= 0 && wx < IMG_W)
                    val = x[((bImg * C_IN + ci) << 12) + (hy << 6) + wx];
            }
            if (j == 0) Bv.x = val; else Bv.y = val;
        }

        // D = A*B + C  (f32, 16x16x4)
        acc = __builtin_amdgcn_wmma_f32_16x16x4_f32(
                  false, A, false, Bv, (short)0, acc, false, false);
    }

    // ---- spike expansion + streamed one-hot NT stores ---------------------
    // C/D layout: VGPR r holds row M=r (lanes 0-15) / M=r+8 (lanes 16-31).
    #pragma unroll
    for (int r = 0; r < 8; ++r) {
        const int   c    = chBase + r + hiHalf * 8;
        const float conv = acc[r];
        const float iv   = invn[c];
        const float bb   = bsh[c];

        float* dst = out + ((size_t)(bImg * C_OUT + c) * T_STEPS) * HW
                         + (size_t)((h << 6) + wc);

        float mem   = 0.f;
        int   fired = 0;
        #pragma unroll 4
        for (int t = 0; t < T_STEPS; ++t) {
            mem += conv;                          // sequential cumsum, matches ref
            const int s = (!fired) && (mem * iv - bb > 0.f);
            // write-once 256 MB stream > 192 MB L2: non-temporal store
            __builtin_nontemporal_store(s ? 1.0f : 0.0f, dst + (size_t)t * HW);
            if (s) { atomicAdd(&cnt[t], 1u); fired = 1; }
        }
    }

    __syncthreads();
    if (tid < T_STEPS) atomicAdd(&gcnt[tid], cnt[tid]);
}

__global__ void zero_counters_kernel(unsigned int* __restrict__ g)
{
    if (threadIdx.x < T_STEPS) g[threadIdx.x] = 0u;
}

__global__ void finalize_kernel(const unsigned int* __restrict__ g,
                                float* __restrict__ out)
{
    if (threadIdx.x == 0) {
        unsigned long long total = 0ull;
        unsigned int mx = 0u;
        for (int t = 0; t < T_STEPS; ++t) {
            total += g[t];
            if (g[t] > mx) mx = g[t];
        }
        out[SPK_ELEMS]     = 0.5f * ((float)total / (float)SPK_ELEMS); // loss
        out[SPK_ELEMS + 1] = (float)mx / PIX_DENOM;                    // spread
    }
}

extern "C" void kernel_launch(void* const* d_in, const int* in_sizes, int n_in,
                              void* d_out, int out_size, void* d_ws, size_t ws_size,
                              hipStream_t stream)
{
    const float* x  = (const float*)d_in[0];   // (16,3,64,64)
    const float* w  = (const float*)d_in[1];   // (32,3,3,3)
    const float* b  = (const float*)d_in[2];   // (32,)
    // d_in[3] = sigma: only used by the custom VJP backward, not the forward.
    float*        out  = (float*)d_out;
    unsigned int* gcnt = (unsigned int*)d_ws;  // 32 spike counters

    zero_counters_kernel<<<1, 32, 0, stream>>>(gcnt);
    // 8192 wave-tiles (4096 pixel tiles x 2 channel tiles), 8 waves per block
    spiking_conv_wmma_kernel<<<1024, 256, 0, stream>>>(x, w, b, out, gcnt);
    finalize_kernel<<<1, 32, 0, stream>>>(gcnt, out);
}